// HistogramMatcher_14534169330278
// MI455X (gfx1250) — compile-verified
//
#include <hip/hip_runtime.h>

#define NBINS 256

typedef __attribute__((ext_vector_type(2))) float v2f;
typedef __attribute__((ext_vector_type(8))) float v8f;

static __device__ __forceinline__ float rescale01(float v) {
    return (v + 1.0f) * (127.0f / 255.0f);
}

// floating_space[i] = clip(i * float32(1/255), 0, 1): only i==255 clips (255*s = 1.0000001 -> 1)
static __device__ __forceinline__ float fspace(int i) {
    float v = (float)i * (1.0f / 255.0f);
    return v > 1.0f ? 1.0f : v;
}

static __device__ __forceinline__ int bin_of(float x) {
    int idx = (int)(x * 256.0f);          // x >= 0, trunc == floor (matches astype(int32))
    idx = idx < 0 ? 0 : idx;
    return idx > (NBINS - 1) ? (NBINS - 1) : idx;
}

// ---------------------------------------------------------------- zero bins
__global__ void hm_zero(unsigned* __restrict__ bins) {
    bins[blockIdx.x * blockDim.x + threadIdx.x] = 0u;
}

// ---------------------------------------------------------------- histograms
__global__ __launch_bounds__(256) void hm_hist(const float* __restrict__ src,
                                               const float* __restrict__ tgt,
                                               unsigned* __restrict__ hist_src,
                                               unsigned* __restrict__ hist_tgt,
                                               int npix) {
    __shared__ unsigned hs[NBINS];
    __shared__ unsigned ht[NBINS];
    const int tid = threadIdx.x;
    hs[tid] = 0u;
    ht[tid] = 0u;
    __syncthreads();

    const int groups = npix >> 2;                  // 4 pixels (12 floats = 3 x b128) per thread
    const int g = blockIdx.x * blockDim.x + tid;
    if (g < groups) {
        const float4* s4 = (const float4*)src;
        const float4* t4 = (const float4*)tgt;
        float4 s0 = s4[3 * g + 0], s1 = s4[3 * g + 1], s2 = s4[3 * g + 2];
        float4 t0 = t4[3 * g + 0], t1 = t4[3 * g + 1], t2 = t4[3 * g + 2];

        float sc[4] = { s0.x, s0.w, s1.z, s2.y };                 // src channel 0
        float tr[4] = { t0.x, t0.w, t1.z, t2.y };
        float tg[4] = { t0.y, t1.x, t1.w, t2.z };
        float tb[4] = { t0.z, t1.y, t2.x, t2.w };
#pragma unroll
        for (int p = 0; p < 4; ++p) {
            atomicAdd(&hs[bin_of(rescale01(sc[p]))], 1u);
            float y = rescale01(tr[p]) * 0.299f + rescale01(tg[p]) * 0.587f +
                      rescale01(tb[p]) * 0.114f;
            atomicAdd(&ht[bin_of(y)], 1u);
        }
    }
    if (blockIdx.x == 0 && tid == 0) {             // scalar tail (npix % 4)
        for (int p = groups << 2; p < npix; ++p) {
            atomicAdd(&hs[bin_of(rescale01(src[3 * p]))], 1u);
            float y = rescale01(tgt[3 * p]) * 0.299f + rescale01(tgt[3 * p + 1]) * 0.587f +
                      rescale01(tgt[3 * p + 2]) * 0.114f;
            atomicAdd(&ht[bin_of(y)], 1u);
        }
    }
    __syncthreads();
    unsigned c = hs[tid];
    if (c) atomicAdd(&hist_src[tid], c);
    c = ht[tid];
    if (c) atomicAdd(&hist_tgt[tid], c);
}

// ---------------------------------------------------------------- CDF + table (1 wave)
// cumsum(hist) over the 16x16 view H:  cdf = H*U + (Lstrict*H)*J, all exact in f32
// (counts <= 2^20 < 2^24). 12x V_WMMA_F32_16X16X4_F32.
static __device__ __forceinline__ v8f wmma4(v2f a, v2f b, v8f c) {
    return __builtin_amdgcn_wmma_f32_16x16x4_f32(false, a, false, b, (short)0, c,
                                                 false, false);
}

__global__ __launch_bounds__(32) void hm_tables(const unsigned* __restrict__ hist_src,
                                                const unsigned* __restrict__ hist_tgt,
                                                float* __restrict__ pxmap,
                                                int npix) {
    __shared__ float H[256];      // current histogram as 16x16 row-major f32
    __shared__ float T[256];      // Lstrict*H staging (C-layout -> row-major)
    __shared__ float CDFS[256];
    __shared__ float CDFT[256];

    const int lane = threadIdx.x;
    const int half = lane >> 4;   // 0: lanes 0-15, 1: lanes 16-31
    const int lm   = lane & 15;
    const float den = (float)(npix - 1);

    for (int pass = 0; pass < 2; ++pass) {
        const unsigned* hist = pass ? hist_tgt : hist_src;
        float* CDF = pass ? CDFT : CDFS;

#pragma unroll
        for (int j = 0; j < 8; ++j)
            H[lane * 8 + j] = (float)hist[lane * 8 + j];
        __syncthreads();

        v8f c0 = {0.f, 0.f, 0.f, 0.f, 0.f, 0.f, 0.f, 0.f};   // H*U  (row prefix)
        v8f t1 = {0.f, 0.f, 0.f, 0.f, 0.f, 0.f, 0.f, 0.f};   // Lstrict*H
#pragma unroll
        for (int kk = 0; kk < 4; ++kk) {
            const int k0 = 4 * kk + 2 * half;                 // A/B: K = k0, k0+1 this half
            v2f a, b;
            a.x = H[16 * lm + k0];                            // A[m][k] = H[m][k]
            a.y = H[16 * lm + k0 + 1];
            b.x = (k0 <= lm) ? 1.0f : 0.0f;                   // U[k][n] = (k <= n)
            b.y = (k0 + 1 <= lm) ? 1.0f : 0.0f;
            c0 = wmma4(a, b, c0);

            v2f a2, b2;
            a2.x = (k0 < lm) ? 1.0f : 0.0f;                   // Lstrict[m][k] = (k < m)
            a2.y = (k0 + 1 < lm) ? 1.0f : 0.0f;
            b2.x = H[16 * k0 + lm];                           // B[k][n] = H[k][n]
            b2.y = H[16 * (k0 + 1) + lm];
            t1 = wmma4(a2, b2, t1);
        }
#pragma unroll
        for (int v = 0; v < 8; ++v)                           // C-layout: row v+8*half, col lm
            T[(v + 8 * half) * 16 + lm] = t1[v];
        __syncthreads();

#pragma unroll
        for (int kk = 0; kk < 4; ++kk) {                      // c0 += T1 * J (row offsets)
            const int k0 = 4 * kk + 2 * half;
            v2f a, b;
            a.x = T[16 * lm + k0];
            a.y = T[16 * lm + k0 + 1];
            b.x = 1.0f;
            b.y = 1.0f;
            c0 = wmma4(a, b, c0);
        }

        const float cmin = H[0];                              // min(cdf) == cdf[0] == hist[0]
#pragma unroll
        for (int v = 0; v < 8; ++v)
            CDF[(v + 8 * half) * 16 + lm] = (c0[v] - cmin) / den;
        __syncthreads();
    }

    // pxmap[i] = interpolate(dx=cdftgt, dy=floating_space, x=cdfsrc[i]) — exact argmax(sign) semantics
    for (int j = 0; j < 8; ++j) {
        const int i = lane * 8 + j;
        const float x = CDFS[i];
        int first_gt = -1, first_eq = -1;
        for (int q = 0; q < NBINS; ++q) {
            float d = CDFT[q];
            if (first_gt < 0 && d > x) first_gt = q;
            if (first_eq < 0 && d == x) first_eq = q;
        }
        int i1 = (first_gt >= 0) ? first_gt : ((first_eq >= 0) ? first_eq : 0);
        int i0 = i1 - 1;
        i0 = i0 < 0 ? 0 : (i0 > NBINS - 1 ? NBINS - 1 : i0);
        float dx0 = CDFT[i0], dx1 = CDFT[i1];
        float dy0 = fspace(i0), dy1 = fspace(i1);
        float d = dx1 - dx0;
        float safe = (d == 0.0f) ? 1.0f : d;
        float interp = dy0 + (dy1 - dy0) * (x - dx0) / safe;
        float res = (x <= CDFT[0]) ? fspace(0)
                  : ((x >= CDFT[255]) ? fspace(255) : interp);
        pxmap[i] = res;
    }
}

// ---------------------------------------------------------------- final per-pixel map
static __device__ __forceinline__ float map_one(float x, const float* pm) {
    float res;
    if (x <= 0.0f) {
        res = pm[0];
    } else if (x >= 1.0f) {
        res = pm[255];
    } else {
        // first j with fspace(j) > x  (exists: fspace(255)=1 > x); boundary-exact fixup
        int j1 = (int)(x * 255.0f);
        j1 = j1 < 0 ? 0 : (j1 > 255 ? 255 : j1);
        while (j1 > 0 && fspace(j1) > x) --j1;
        while (j1 < 255 && fspace(j1) <= x) ++j1;
        const int j0 = j1 - 1;
        const float dx0 = fspace(j0), dx1 = fspace(j1);
        res = pm[j0] + (pm[j1] - pm[j0]) * (x - dx0) / (dx1 - dx0);
    }
    res = res < 0.0f ? 0.0f : (res > 1.0f ? 1.0f : res);
    return res * (255.0f / 127.0f) - 1.0f;
}

__global__ __launch_bounds__(256) void hm_map(const float* __restrict__ src,
                                              const float* __restrict__ pxmap,
                                              float* __restrict__ out,
                                              int npix) {
    __shared__ float pm[NBINS];
    const int tid = threadIdx.x;
    pm[tid] = pxmap[tid];
    __syncthreads();

    const int groups = npix >> 2;
    const int g = blockIdx.x * blockDim.x + tid;
    if (g < groups) {
        const float4* s4 = (const float4*)src;
        float4 s0 = s4[3 * g + 0], s1 = s4[3 * g + 1], s2 = s4[3 * g + 2];
        float sc[4] = { s0.x, s0.w, s1.z, s2.y };
        float4 o;
        o.x = map_one(rescale01(sc[0]), pm);
        o.y = map_one(rescale01(sc[1]), pm);
        o.z = map_one(rescale01(sc[2]), pm);
        o.w = map_one(rescale01(sc[3]), pm);
        ((float4*)out)[g] = o;
    }
    if (blockIdx.x == 0 && tid == 0) {
        for (int p = groups << 2; p < npix; ++p)
            out[p] = map_one(rescale01(src[3 * p]), pm);
    }
}

// ---------------------------------------------------------------- launch
extern "C" void kernel_launch(void* const* d_in, const int* in_sizes, int n_in,
                              void* d_out, int out_size, void* d_ws, size_t ws_size,
                              hipStream_t stream) {
    const float* src = (const float*)d_in[0];
    const float* tgt = (const float*)d_in[1];
    float* out = (float*)d_out;
    const int npix = in_sizes[0] / 3;

    unsigned* hist_src = (unsigned*)d_ws;          // [0..255]
    unsigned* hist_tgt = hist_src + NBINS;         // [256..511]
    float* pxmap = (float*)(hist_tgt + NBINS);     // [512..767] (as float)

    hm_zero<<<2, 256, 0, stream>>>(hist_src);      // zeroes both histograms (512 u32)

    const int groups = npix >> 2;
    int blocks = (groups + 255) / 256;
    if (blocks < 1) blocks = 1;

    hm_hist<<<blocks, 256, 0, stream>>>(src, tgt, hist_src, hist_tgt, npix);
    hm_tables<<<1, 32, 0, stream>>>(hist_src, hist_tgt, pxmap, npix);
    hm_map<<<blocks, 256, 0, stream>>>(src, pxmap, out, npix);
}